// CentDif_8787503087639
// MI455X (gfx1250) — compile-verified
//
#include <hip/hip_runtime.h>
#include <cstdint>

typedef float v4f __attribute__((ext_vector_type(4)));

// Problem constants (from the reference)
constexpr int   LDIM  = 2048;
constexpr int   IGST  = 10;
constexpr int   TILE  = 1024;               // l-positions handled per block
constexpr int   HALO  = 16;                 // halo floats each side (b128-aligned)
constexpr int   REGION = TILE + 2 * HALO;   // 1056 floats per channel in LDS
constexpr int   CHUNKS = REGION / 4;        // 264 b128 chunks per channel
constexpr int   NCHUNK = 2 * CHUNKS;        // 528 chunks total (2 channels)

// CDNA5 async global->LDS copy, 16B per lane (tracked by ASYNCcnt).
// GV addressing: vdst = LDS byte address, vaddr = 64-bit global address.
__device__ __forceinline__ void async_load_b128(uint32_t lds_byte_addr,
                                                uint64_t global_addr) {
    asm volatile("global_load_async_to_lds_b128 %0, %1, off"
                 :: "v"(lds_byte_addr), "v"(global_addr)
                 : "memory");
}

__device__ __forceinline__ void wait_asynccnt0() {
    asm volatile("s_wait_asynccnt 0" ::: "memory");
}

__global__ __launch_bounds__(256)
void centdif_kernel(const float* __restrict__ u, float* __restrict__ out) {
    __shared__ float smem[2 * REGION];      // 8448 bytes

    const int bid  = blockIdx.x;
    const int m    = bid >> 1;              // LDIM / TILE == 2 tiles per row
    const int base = (bid & 1) * TILE;
    const int tid  = threadIdx.x;

    const size_t row = (size_t)(2 * m) * LDIM;   // offset of u[m][0][0]

    // ---- Stage [base-HALO, base+TILE+HALO) for both channels into LDS ----
    for (int chunk = tid; chunk < NCHUNK; chunk += 256) {
        const int n = chunk / CHUNKS;            // channel
        const int i = chunk - n * CHUNKS;        // b128 chunk within channel
        int s = base - HALO + 4 * i;             // global l of chunk start
        s = s < 0 ? 0 : (s > LDIM - 4 ? LDIM - 4 : s);  // clamp (never tapped)
        const float* gp = u + row + (size_t)n * LDIM + (size_t)s;
        const uint32_t lb = (uint32_t)(uintptr_t)(&smem[n * REGION + 4 * i]);
        async_load_b128(lb, (uint64_t)(uintptr_t)gp);
    }
    wait_asynccnt0();       // my wave's async copies landed in LDS
    __syncthreads();        // everyone's copies visible

    // ---- Compute: 4 consecutive l per thread, both channels ----
    const float c = (float)(1.0 / (60.0 * 0.012));   // 1/(60*DX)

    const v4f* sv0 = (const v4f*)(&smem[0]);
    const v4f* sv1 = (const v4f*)(&smem[REGION]);

    // 12-float sliding window per channel: region floats [4t+12, 4t+24)
    v4f a0 = sv0[tid + 3], b0 = sv0[tid + 4], c0 = sv0[tid + 5];
    v4f a1 = sv1[tid + 3], b1 = sv1[tid + 4], c1 = sv1[tid + 5];
    const float w0[12] = {a0.x, a0.y, a0.z, a0.w,  b0.x, b0.y, b0.z, b0.w,
                          c0.x, c0.y, c0.z, c0.w};
    const float w1[12] = {a1.x, a1.y, a1.z, a1.w,  b1.x, b1.y, b1.z, b1.w,
                          c1.x, c1.y, c1.z, c1.w};

    float r0[4], r1[4];
#pragma unroll
    for (int j = 0; j < 4; ++j) {
        const int  l        = base + 4 * tid + j;
        const bool interior = (l >= IGST) && (l < LDIM - IGST);
        // w[k] holds global l = base + 4*tid - 4 + k ; tap l+d -> k = j+4+d
        float d0 = interior
                 ? c * (-w0[j + 1] + 9.0f * w0[j + 2] - 45.0f * w0[j + 3]
                        + 45.0f * w0[j + 5] - 9.0f * w0[j + 6] + w0[j + 7])
                 : 0.0f;
        float d1 = interior
                 ? c * (-w1[j + 1] + 9.0f * w1[j + 2] - 45.0f * w1[j + 3]
                        + 45.0f * w1[j + 5] - 9.0f * w1[j + 6] + w1[j + 7])
                 : 0.0f;
        const float u0 = w0[j + 4];
        const float u1 = w1[j + 4];
        r0[j] = -(u1 * d0 + u0 * d1);        // -adu0
        r1[j] = -(2.0f * d0 + u1 * d1);      // -adu1
    }

    const v4f o0 = {r0[0], r0[1], r0[2], r0[3]};
    const v4f o1 = {r1[0], r1[1], r1[2], r1[3]};

    // Streaming (non-temporal) b128 stores: output is write-once, keep L2 clean.
    v4f* p0 = (v4f*)(out + row + (size_t)base + 4 * tid);
    v4f* p1 = (v4f*)(out + row + (size_t)LDIM + (size_t)base + 4 * tid);
    __builtin_nontemporal_store(o0, p0);
    __builtin_nontemporal_store(o1, p1);
}

extern "C" void kernel_launch(void* const* d_in, const int* in_sizes, int n_in,
                              void* d_out, int out_size, void* d_ws, size_t ws_size,
                              hipStream_t stream) {
    const float* u = (const float*)d_in[0];
    float* out = (float*)d_out;
    const int M = in_sizes[0] / (2 * LDIM);       // 16384
    const int blocks = M * (LDIM / TILE);         // 32768
    centdif_kernel<<<dim3(blocks), dim3(256), 0, stream>>>(u, out);
}